// kernel_generated_2_21887153341384
// MI455X (gfx1250) — compile-verified
//
#include <hip/hip_runtime.h>
#include <hip/hip_bf16.h>

// ---------------------------------------------------------------------------
// out[0, n*96+j, l, m] = sum_{g<24} A[n,j,g] * x[0, n*24+g, m, l]
//   where A[n,j,g] = sum_{i<9} w0[j,i] * w1[n*24+g, i]
//
// Memory-bound (reads 602KB, writes 2.4MB). Fused single pass over x/out
// using V_WMMA_F32_16X16X4_F32 (full fp32 precision on the matrix pipe).
// ---------------------------------------------------------------------------

typedef __attribute__((ext_vector_type(2))) float v2f;
typedef __attribute__((ext_vector_type(8))) float v8f;

#define SPATIAL 3136      // 56*56
#define N_GROUPS 2
#define JTILES 6          // 96 / 16
#define STILES 196        // 3136 / 16
#define TILES_TOTAL (N_GROUPS * JTILES * STILES)   // 2352
#define WAVES_PER_BLOCK 8
#define BLOCKS (TILES_TOTAL / WAVES_PER_BLOCK)     // 294

// Kernel 1: fold the i=9 contraction: A[((n*96)+j)*24 + g] = <w0[j,:], w1[n*24+g,:]>
__global__ void build_A_kernel(const float* __restrict__ w0,
                               const float* __restrict__ w1,
                               float* __restrict__ A) {
    int idx = blockIdx.x * blockDim.x + threadIdx.x;
    if (idx >= N_GROUPS * 96 * 24) return;
    int g = idx % 24;
    int t = idx / 24;
    int j = t % 96;
    int n = t / 96;
    const float* w0r = w0 + j * 9;
    const float* w1r = w1 + (n * 24 + g) * 9;
    float acc = 0.f;
#pragma unroll
    for (int i = 0; i < 9; ++i) acc += w0r[i] * w1r[i];
    A[idx] = acc;
}

// Kernel 2: per-wave 16x16 output tile, K=24 via 6 x wmma_f32_16x16x4_f32.
__global__ __launch_bounds__(WAVES_PER_BLOCK * 32)
void fused_wmma_kernel(const float* __restrict__ x,
                       const float* __restrict__ A,
                       float* __restrict__ out) {
    const int wid  = (blockIdx.x * blockDim.x + threadIdx.x) >> 5;  // wave-uniform
    const int lane = threadIdx.x & 31;
    if (wid >= TILES_TOTAL) return;   // uniform per wave; exact grid anyway

    const int n  = wid / (JTILES * STILES);
    const int r0 = wid % (JTILES * STILES);
    const int jt = r0 / STILES;
    const int st = r0 % STILES;
    const int j0 = jt << 4;
    const int s0 = st << 4;

    const int lane15 = lane & 15;
    const int khi    = lane >> 4;     // 0 or 1

    // B addressing: output col s -> x spatial index m*56 + l (transposed)
    const int s = s0 + lane15;
    const int l = s / 56;
    const int m = s - l * 56;
    const float* xb = x + (size_t)n * 24 * SPATIAL + (size_t)m * 56 + l;

    // A addressing: row j0+lane15, k offset 2*khi within each K=4 chunk
    const float* ab = A + ((size_t)(n * 96 + j0 + lane15)) * 24 + 2 * khi;

    v8f c = {};
#pragma unroll
    for (int kk = 0; kk < 6; ++kk) {
        // A frag (16x4): vgpr0 = K {0,2}, vgpr1 = K {1,3} (consecutive per lane)
        v2f a = *(const v2f*)(ab + kk * 4);
        // B frag (4x16): row 2*khi in vgpr0, row 2*khi+1 in vgpr1
        const int ch = kk * 4 + 2 * khi;
        v2f b;
        b.x = xb[(size_t)ch * SPATIAL];
        b.y = xb[(size_t)(ch + 1) * SPATIAL];
        // 8 args: (neg_a, A, neg_b, B, c_mod, C, reuse_a, reuse_b)
        c = __builtin_amdgcn_wmma_f32_16x16x4_f32(false, a, false, b,
                                                  (short)0, c, false, false);
    }

    // C/D layout: vgpr r -> row j0 + 8*khi + r, col s0 + lane15
    float* ob = out + ((size_t)(n * 96 + j0 + 8 * khi)) * SPATIAL + s;
#pragma unroll
    for (int rr = 0; rr < 8; ++rr)
        ob[(size_t)rr * SPATIAL] = c[rr];
}

extern "C" void kernel_launch(void* const* d_in, const int* in_sizes, int n_in,
                              void* d_out, int out_size, void* d_ws, size_t ws_size,
                              hipStream_t stream) {
    const float* x  = (const float*)d_in[0];   // (1, 48, 56, 56)
    const float* w0 = (const float*)d_in[1];   // (96, 9)
    const float* w1 = (const float*)d_in[2];   // (48, 9)
    float* out = (float*)d_out;                // (1, 192, 56, 56)
    float* A   = (float*)d_ws;                 // 2*96*24 floats = 18 KB

    build_A_kernel<<<(N_GROUPS * 96 * 24 + 127) / 128, 128, 0, stream>>>(w0, w1, A);
    fused_wmma_kernel<<<BLOCKS, WAVES_PER_BLOCK * 32, 0, stream>>>(x, A, out);
}